// HybridAttentionCritic_17197049053797
// MI455X (gfx1250) — compile-verified
//
#include <hip/hip_runtime.h>
#include <hip/hip_bf16.h>

typedef __attribute__((ext_vector_type(2))) float v2f;
typedef __attribute__((ext_vector_type(8))) float v8f;

constexpr int AG_IN  = 64;    // input features per agent
constexpr int EMB    = 128;   // embed size
constexpr int HID1   = 512;
constexpr int HID2   = 256;
constexpr int TILE_M = 128;   // agents per block
constexpr int NWAVES = 8;

// workspace layout (floats)
constexpr int WS_ESELF    = 0;    // [128] embeds[self]
constexpr int WS_G        = 128;  // [128] g = Wk^T @ q
constexpr int WS_C        = 256;  // [1]   c = bk . q
constexpr int WS_WEIGHTED = 320;  // [128] accumulated weighted sum

// ---------------------------------------------------------------------------
// Stage 1: per-call setup. embed_self, q, g = Wk^T q, c = bk.q ; zero accum.
// ---------------------------------------------------------------------------
__global__ void hac_stage1(const float* __restrict__ obs,
                           const float* __restrict__ We, const float* __restrict__ be,
                           const float* __restrict__ Wq, const float* __restrict__ bq,
                           const float* __restrict__ Wk, const float* __restrict__ bk,
                           const int* __restrict__ self_idx,
                           float* __restrict__ ws) {
  __shared__ float s_es[EMB], s_q[EMB];
  const int tid = threadIdx.x;          // 128 threads
  const int self = self_idx[0];
  const float* orow = obs + (size_t)self * AG_IN;
  float acc = be[tid];
  for (int k = 0; k < AG_IN; ++k) acc += orow[k] * We[tid * AG_IN + k];
  s_es[tid] = acc;
  ws[WS_ESELF + tid] = acc;
  __syncthreads();
  float q = bq[tid];
  for (int e = 0; e < EMB; ++e) q += s_es[e] * Wq[tid * EMB + e];
  s_q[tid] = q;
  __syncthreads();
  float g = 0.f;
  for (int j = 0; j < EMB; ++j) g += s_q[j] * Wk[j * EMB + tid];
  ws[WS_G + tid] = g;
  ws[WS_WEIGHTED + tid] = 0.f;
  if (tid == 0) {
    float c = 0.f;
    for (int j = 0; j < EMB; ++j) c += s_q[j] * bk[j];
    ws[WS_C] = c;
  }
}

// ---------------------------------------------------------------------------
// Cooperative transposed tile load: 64(n) x 64(k) f32 tile from row-major
// global (row n, stride `rowstride`) into paired-K LDS staging so a WMMA
// B-fragment (B[k][n], B[k+1][n]) is one ds_load_b64.
//   dst[k/2][n] = { src[n][k], src[n][k+1] }
// Must be called by all 256 threads.
// ---------------------------------------------------------------------------
__device__ __forceinline__ void load_tileT(const float* __restrict__ src,
                                           int rowstride,
                                           float2 (*__restrict__ dst)[68],
                                           int tid) {
  #pragma unroll
  for (int rep = 0; rep < 4; ++rep) {
    const int n  = (tid >> 4) + rep * 16;
    const int k4 = (tid & 15) << 2;
    const float4 v = *(const float4*)(src + (size_t)n * rowstride + k4);
    dst[(k4 >> 1) + 0][n] = make_float2(v.x, v.y);
    dst[(k4 >> 1) + 1][n] = make_float2(v.z, v.w);
  }
}

// ---------------------------------------------------------------------------
// Stage 2: fused embed GEMM + scores + scalar-input attention MLP + softmax
//          + values GEMM + masked weighted reduction. 8 waves, 16 agents/wave.
// All matrix math on v_wmma_f32_16x16x4_f32; all B-matrices staged in LDS.
// fp32 WMMA fragment layouts (ISA 7.12.2):
//   A 16x4 : lane L -> row M=L%16, a[v] = A[M][2*(L/16)+v]
//   B 4x16 : lane L -> col N=L%16, b[v] = B[2*(L/16)+v][N]
//   C 16x16: lane L -> col N=L%16, c[r] = C[r+8*(L/16)][N]
// ---------------------------------------------------------------------------
__launch_bounds__(256, 1)
__global__ void hac_stage2(const float* __restrict__ obs,
                           const float* __restrict__ We, const float* __restrict__ be,
                           const float* __restrict__ Wv,
                           const float* __restrict__ W1, const float* __restrict__ b1,
                           const float* __restrict__ W2, const float* __restrict__ b2,
                           const float* __restrict__ W3, const float* __restrict__ b3,
                           const int* __restrict__ self_idx,
                           float* __restrict__ ws, int N) {
  __shared__ float  s_obs[TILE_M][AG_IN + 4];   // 34.8 KB, stride%64==4 -> conflict-free frags
  __shared__ float  s_emb[TILE_M][EMB + 4];     // 67.6 KB
  __shared__ float  s_w1[HID1], s_b1[HID1];     //  4.0 KB
  __shared__ float  s_g[EMB];
  __shared__ float  s_scores[TILE_M];
  __shared__ float  s_hstage[NWAVES][16][68];   // 34.8 KB per-wave h2 transpose stage
  __shared__ float2 s_bstage[32][68];           // 17.4 KB shared B-matrix chunk (64k x 64n)
  __shared__ float  s_weighted[EMB];

  const int tid  = threadIdx.x;
  const int wave = tid >> 5;
  const int lane = tid & 31;
  const int hh   = lane >> 4;     // half-wave select
  const int mr   = lane & 15;
  const int rowbase = wave * 16;  // this wave's 16 agents within the tile
  const int self   = self_idx[0];
  const int ablock = blockIdx.x * TILE_M;

  // ---- phase 1: cooperative loads ----
  for (int i = tid; i < TILE_M * (AG_IN / 4); i += 256) {
    int r  = i >> 4;
    int c4 = (i & 15) << 2;
    int a  = ablock + r;
    float4 v = make_float4(0.f, 0.f, 0.f, 0.f);
    if (a < N) v = *(const float4*)(obs + (size_t)a * AG_IN + c4);
    s_obs[r][c4 + 0] = v.x; s_obs[r][c4 + 1] = v.y;
    s_obs[r][c4 + 2] = v.z; s_obs[r][c4 + 3] = v.w;
  }
  for (int i = tid; i < HID1; i += 256) { s_w1[i] = W1[i]; s_b1[i] = b1[i]; }
  if (tid < EMB) { s_g[tid] = ws[WS_G + tid]; s_weighted[tid] = 0.f; }

  // ---- phase 2: GEMM1  emb = obs @ We^T + be  (B staged in LDS) ----
  for (int nh = 0; nh < 2; ++nh) {
    __syncthreads();
    load_tileT(We + (size_t)(nh * 64) * AG_IN, AG_IN, s_bstage, tid);
    __syncthreads();
    #pragma unroll
    for (int t2 = 0; t2 < 4; ++t2) {
      const int t = nh * 4 + t2;
      const int nloc = t2 * 16 + mr;
      const float bv = be[t * 16 + mr];
      v8f acc = {bv, bv, bv, bv, bv, bv, bv, bv};
      #pragma unroll
      for (int k0 = 0; k0 < AG_IN; k0 += 4) {
        v2f a; a.x = s_obs[rowbase + mr][k0 + 2 * hh];
               a.y = s_obs[rowbase + mr][k0 + 2 * hh + 1];
        const float2 bb = s_bstage[(k0 >> 1) + hh][nloc];
        v2f b; b.x = bb.x; b.y = bb.y;
        acc = __builtin_amdgcn_wmma_f32_16x16x4_f32(false, a, false, b, (short)0, acc,
                                                    false, false);
      }
      #pragma unroll
      for (int r = 0; r < 8; ++r)
        s_emb[rowbase + r + 8 * hh][t * 16 + mr] = acc[r];
    }
  }
  __syncthreads();

  // ---- phase 3: scores_i = (emb_i . g + c) / sqrt(E) ----
  if (tid < TILE_M) {
    float sc = ws[WS_C];
    for (int e = 0; e < EMB; ++e) sc += s_emb[tid][e] * s_g[e];
    s_scores[tid] = sc * 0.0883883476483184f;   // 1/sqrt(128)
  }
  __syncthreads();

  // ---- phase 4: scalar-input MLP  (h1 generated on the fly; h2 via WMMA with
  //               LDS-staged W2 chunks; h2 transposed through per-wave LDS;
  //               sim accumulated with LDS-staged W3 chunks) ----
  const float sm = s_scores[rowbase + mr];
  v8f sim[8];
  #pragma unroll
  for (int t = 0; t < 8; ++t) {
    const float bv = b3[t * 16 + mr];
    v8f tmp = {bv, bv, bv, bv, bv, bv, bv, bv};
    sim[t] = tmp;
  }
  for (int nc = 0; nc < 4; ++nc) {              // 64-wide chunks of h2's 256 cols
    v8f acc2[4];
    #pragma unroll
    for (int tt = 0; tt < 4; ++tt) {
      const float bv = b2[nc * 64 + tt * 16 + mr];
      v8f tmp = {bv, bv, bv, bv, bv, bv, bv, bv};
      acc2[tt] = tmp;
    }
    for (int kc = 0; kc < 8; ++kc) {            // K chunks of 64 over HID1=512
      __syncthreads();
      load_tileT(W2 + (size_t)(nc * 64) * HID1 + kc * 64, HID1, s_bstage, tid);
      if (kc < 7 && tid < 64)                   // prefetch next K-chunk rows
        __builtin_prefetch(W2 + (size_t)(nc * 64 + tid) * HID1 + (kc + 1) * 64);
      __syncthreads();
      #pragma unroll
      for (int tt = 0; tt < 4; ++tt) {
        v8f acc = acc2[tt];
        #pragma unroll
        for (int k0 = 0; k0 < 64; k0 += 4) {
          const int kg = kc * 64 + k0 + 2 * hh;
          v2f a; a.x = fmaxf(fmaf(sm, s_w1[kg],     s_b1[kg]),     0.f);
                 a.y = fmaxf(fmaf(sm, s_w1[kg + 1], s_b1[kg + 1]), 0.f);
          const float2 bb = s_bstage[(k0 >> 1) + hh][tt * 16 + mr];
          v2f b; b.x = bb.x; b.y = bb.y;
          acc = __builtin_amdgcn_wmma_f32_16x16x4_f32(false, a, false, b, (short)0, acc,
                                                      false, false);
        }
        acc2[tt] = acc;
      }
    }
    // stage relu(h2 chunk) transposed for use as A in the W3 GEMM
    #pragma unroll
    for (int tt = 0; tt < 4; ++tt)
      #pragma unroll
      for (int r = 0; r < 8; ++r)
        s_hstage[wave][r + 8 * hh][tt * 16 + mr] = fmaxf(acc2[tt][r], 0.f);
    asm volatile("s_wait_dscnt 0" ::: "memory");   // wave-private stage: store->load
    for (int th = 0; th < 2; ++th) {               // sim cols in halves of 64
      __syncthreads();
      load_tileT(W3 + (size_t)(th * 64) * HID2 + nc * 64, HID2, s_bstage, tid);
      __syncthreads();
      #pragma unroll
      for (int t2 = 0; t2 < 4; ++t2) {
        const int t = th * 4 + t2;
        v8f acc = sim[t];
        #pragma unroll
        for (int k0 = 0; k0 < 64; k0 += 4) {
          const int kk = k0 + 2 * hh;
          v2f a; a.x = s_hstage[wave][mr][kk]; a.y = s_hstage[wave][mr][kk + 1];
          const float2 bb = s_bstage[(k0 >> 1) + hh][t2 * 16 + mr];
          v2f b; b.x = bb.x; b.y = bb.y;
          acc = __builtin_amdgcn_wmma_f32_16x16x4_f32(false, a, false, b, (short)0, acc,
                                                      false, false);
        }
        sim[t] = acc;
      }
    }
  }

  // ---- phase 5: relu + row softmax (over 128 features) + validity mask ----
  #pragma unroll
  for (int r = 0; r < 8; ++r) {
    float m = -1e30f;
    #pragma unroll
    for (int t = 0; t < 8; ++t) {
      const float v = fmaxf(sim[t][r], 0.f);
      sim[t][r] = v;
      m = fmaxf(m, v);
    }
    m = fmaxf(m, __shfl_xor(m, 1));
    m = fmaxf(m, __shfl_xor(m, 2));
    m = fmaxf(m, __shfl_xor(m, 4));
    m = fmaxf(m, __shfl_xor(m, 8));
    float s = 0.f;
    #pragma unroll
    for (int t = 0; t < 8; ++t) {
      const float e = __expf(sim[t][r] - m);
      sim[t][r] = e;
      s += e;
    }
    s += __shfl_xor(s, 1);
    s += __shfl_xor(s, 2);
    s += __shfl_xor(s, 4);
    s += __shfl_xor(s, 8);
    const int agent = ablock + rowbase + r + 8 * hh;
    const float scale = (agent < N && agent != self) ? (1.f / s) : 0.f;
    #pragma unroll
    for (int t = 0; t < 8; ++t) sim[t][r] *= scale;
  }

  // ---- phase 6: values = emb @ Wv^T (LDS-staged Wv) ; acc sum_i w_i*v_i ----
  v8f vacc[8];
  #pragma unroll
  for (int t = 0; t < 8; ++t) {
    v8f z = {0.f, 0.f, 0.f, 0.f, 0.f, 0.f, 0.f, 0.f};
    vacc[t] = z;
  }
  for (int kc = 0; kc < 2; ++kc) {
    for (int nh = 0; nh < 2; ++nh) {
      __syncthreads();
      load_tileT(Wv + (size_t)(nh * 64) * EMB + kc * 64, EMB, s_bstage, tid);
      __syncthreads();
      #pragma unroll
      for (int t2 = 0; t2 < 4; ++t2) {
        const int t = nh * 4 + t2;
        v8f acc = vacc[t];
        #pragma unroll
        for (int k0 = 0; k0 < 64; k0 += 4) {
          const int kg = kc * 64 + k0 + 2 * hh;
          v2f a; a.x = s_emb[rowbase + mr][kg];
                 a.y = s_emb[rowbase + mr][kg + 1];
          const float2 bb = s_bstage[(k0 >> 1) + hh][t2 * 16 + mr];
          v2f b; b.x = bb.x; b.y = bb.y;
          acc = __builtin_amdgcn_wmma_f32_16x16x4_f32(false, a, false, b, (short)0, acc,
                                                      false, false);
        }
        vacc[t] = acc;
      }
    }
  }
  #pragma unroll
  for (int t = 0; t < 8; ++t) {
    float p = 0.f;
    #pragma unroll
    for (int r = 0; r < 8; ++r) p += sim[t][r] * vacc[t][r];
    p += __shfl_xor(p, 16);                     // fold the two 8-agent halves
    if (lane < 16) atomicAdd(&s_weighted[t * 16 + mr], p);
  }
  __syncthreads();
  if (tid < EMB) atomicAdd(&ws[WS_WEIGHTED + tid], s_weighted[tid]);
}

// ---------------------------------------------------------------------------
// Stage 3: critic head on [embed_self ; weighted]
// ---------------------------------------------------------------------------
__global__ void hac_stage3(const float* __restrict__ Wo1, const float* __restrict__ bo1,
                           const float* __restrict__ Wo2, const float* __restrict__ bo2,
                           const float* __restrict__ ws, float* __restrict__ out) {
  __shared__ float s_h[128];
  const int tid = threadIdx.x;  // 128 threads
  float acc = bo1[tid];
  const float* w = Wo1 + (size_t)tid * 256;
  for (int i = 0; i < 128; ++i) acc += ws[WS_ESELF + i] * w[i];
  for (int i = 0; i < 128; ++i) acc += ws[WS_WEIGHTED + i] * w[128 + i];
  s_h[tid] = fmaxf(acc, 0.f) * Wo2[tid];
  __syncthreads();
  for (int s = 64; s > 0; s >>= 1) {
    if (tid < s) s_h[tid] += s_h[tid + s];
    __syncthreads();
  }
  if (tid == 0) out[0] = s_h[0] + bo2[0];
}

// ---------------------------------------------------------------------------
extern "C" void kernel_launch(void* const* d_in, const int* in_sizes, int n_in,
                              void* d_out, int out_size, void* d_ws, size_t ws_size,
                              hipStream_t stream) {
  const float* obs = (const float*)d_in[0];
  const float* We  = (const float*)d_in[1];
  const float* be  = (const float*)d_in[2];
  const float* Wv  = (const float*)d_in[3];
  const float* Wq  = (const float*)d_in[4];
  const float* bq  = (const float*)d_in[5];
  const float* Wk  = (const float*)d_in[6];
  const float* bk  = (const float*)d_in[7];
  const float* W1  = (const float*)d_in[8];
  const float* b1  = (const float*)d_in[9];
  const float* W2  = (const float*)d_in[10];
  const float* b2  = (const float*)d_in[11];
  const float* W3  = (const float*)d_in[12];
  const float* b3  = (const float*)d_in[13];
  const float* Wo1 = (const float*)d_in[14];
  const float* bo1 = (const float*)d_in[15];
  const float* Wo2 = (const float*)d_in[16];
  const float* bo2 = (const float*)d_in[17];
  const int*  self = (const int*)d_in[18];
  const int N = in_sizes[0] / AG_IN;
  float* ws  = (float*)d_ws;
  float* out = (float*)d_out;

  hipLaunchKernelGGL(hac_stage1, dim3(1), dim3(128), 0, stream,
                     obs, We, be, Wq, bq, Wk, bk, self, ws);
  const int blocks = (N + TILE_M - 1) / TILE_M;
  hipLaunchKernelGGL(hac_stage2, dim3(blocks), dim3(256), 0, stream,
                     obs, We, be, Wv, W1, b1, W2, b2, W3, b3, self, ws, N);
  hipLaunchKernelGGL(hac_stage3, dim3(1), dim3(128), 0, stream,
                     Wo1, bo1, Wo2, bo2, ws, out);
}